// GCNEncoder2_53163105190281
// MI455X (gfx1250) — compile-verified
//
#include <hip/hip_runtime.h>

typedef __attribute__((ext_vector_type(2))) float v2f;
typedef __attribute__((ext_vector_type(8))) float v8f;

// ===========================================================================
// Small utility kernels
// ===========================================================================
__global__ void zero_int_kernel(int* __restrict__ p, int n) {
    int i = blockIdx.x * blockDim.x + threadIdx.x;
    if (i < n) p[i] = 0;
}

__global__ void count_kernel(const int* __restrict__ col, int* __restrict__ cnt, int e) {
    int i = blockIdx.x * blockDim.x + threadIdx.x;
    if (i < e) atomicAdd(&cnt[col[i]], 1);
}

// Single-workgroup exclusive scan (N ~ 100k, ~100 chunk iterations -> trivial).
__global__ void scan_kernel(const int* __restrict__ counts, int* __restrict__ offs, int n) {
    __shared__ int buf[1024];
    __shared__ int carrySh;
    const int tid = threadIdx.x;
    if (tid == 0) carrySh = 0;
    __syncthreads();
    for (int base = 0; base < n; base += 1024) {
        int i = base + tid;
        int v = (i < n) ? counts[i] : 0;
        buf[tid] = v;
        __syncthreads();
        for (int d = 1; d < 1024; d <<= 1) {          // Hillis-Steele inclusive
            int t = (tid >= d) ? buf[tid - d] : 0;
            __syncthreads();
            buf[tid] += t;
            __syncthreads();
        }
        int incl  = buf[tid];
        int carry = carrySh;
        if (i < n) offs[i] = carry + incl - v;        // exclusive
        __syncthreads();
        if (tid == 1023) carrySh = carry + incl;
        __syncthreads();
    }
    if (tid == 0) offs[n] = carrySh;
}

// Bucket edges by destination: perm arrays hold (source row, edge attr).
__global__ void permute_kernel(const int* __restrict__ rows, const int* __restrict__ cols,
                               const float* __restrict__ ea,
                               const int* __restrict__ offs, int* __restrict__ cursor,
                               int* __restrict__ permRow, float* __restrict__ permEa, int e) {
    int i = blockIdx.x * blockDim.x + threadIdx.x;
    if (i >= e) return;
    int c = cols[i];
    int pos = offs[c] + atomicAdd(&cursor[c], 1);
    permRow[pos] = rows[i];
    permEa[pos]  = ea[i];
}

__global__ void dinv_from_offs_kernel(const int* __restrict__ offs, float* __restrict__ dinv, int n) {
    int i = blockIdx.x * blockDim.x + threadIdx.x;
    if (i < n) dinv[i] = 1.0f / sqrtf((float)(offs[i + 1] - offs[i] + 1));  // +1 self loop
}

// ---- fallback-path degree kernels (atomic variant) -------------------------
__global__ void deg_init_kernel(float* __restrict__ deg, int n) {
    int i = blockIdx.x * blockDim.x + threadIdx.x;
    if (i < n) deg[i] = 1.0f;
}
__global__ void deg_accum_kernel(const int* __restrict__ col, float* __restrict__ deg, int e) {
    int i = blockIdx.x * blockDim.x + threadIdx.x;
    if (i < e) unsafeAtomicAdd(&deg[col[i]], 1.0f);
}
__global__ void deg_finish_kernel(float* __restrict__ deg, int n) {
    int i = blockIdx.x * blockDim.x + threadIdx.x;
    if (i < n) deg[i] = 1.0f / sqrtf(deg[i]);
}

// ===========================================================================
// Dense transform: out[N, ODIM] = (RELU? max(A,0):A)[N,K] @ W[ODIM,K]^T + bias
// One wave computes a 16x16 tile via V_WMMA_F32_16X16X4_F32 (K steps of 4).
//   A 16x4 : lane<16 -> row M=lane, k-pair {k0,k0+1}; lane>=16 -> {k0+2,k0+3}
//   B 4x16 : lane<16 -> col N=lane, k-pair {k0,k0+1}; lane>=16 -> {k0+2,k0+3}
//   C/D    : VGPR p -> row mBase+p+8*(lane>=16), col nBase+(lane&15)
// ===========================================================================
template<int K, int ODIM, bool RELU>
__global__ void gemm_bias_wmma(const float* __restrict__ A,
                               const float* __restrict__ W,
                               const float* __restrict__ bias,
                               float* __restrict__ out,
                               int mTiles) {
    constexpr int NT = ODIM / 16;
    constexpr int SH = (NT == 8) ? 3 : 2;
    const int wave = threadIdx.x >> 5;
    const int lane = threadIdx.x & 31;
    const int gw   = blockIdx.x * (blockDim.x >> 5) + wave;
    const int mTile = gw >> SH;
    const int nTile = gw & (NT - 1);
    if (mTile >= mTiles) return;            // wave-uniform: EXEC stays all-1s

    const int mBase   = mTile * 16;
    const int nBase   = nTile * 16;
    const int l15     = lane & 15;
    const int halfSel = lane >> 4;
    const int kOff    = halfSel * 2;

    const float* aPtr = A + (size_t)(mBase + l15) * K + kOff;
    const float* wPtr = W + (size_t)(nBase + l15) * K + kOff;

    v8f c;
    const float bv = bias[nBase + l15];
#pragma unroll
    for (int p = 0; p < 8; ++p) c[p] = bv;

#pragma unroll 8
    for (int k0 = 0; k0 < K; k0 += 4) {
        v2f a = *(const v2f*)(aPtr + k0);
        if (RELU) { a.x = fmaxf(a.x, 0.0f); a.y = fmaxf(a.y, 0.0f); }
        v2f b = *(const v2f*)(wPtr + k0);
        c = __builtin_amdgcn_wmma_f32_16x16x4_f32(
                false, a, false, b, (short)0, c, false, false);
    }

    float* oPtr = out + (size_t)(mBase + halfSel * 8) * ODIM + nBase + l15;
#pragma unroll
    for (int p = 0; p < 8; ++p) oPtr[(size_t)p * ODIM] = c[p];
}

template<int K, bool RELU>
__global__ void gemm_tail_kernel(const float* __restrict__ A,
                                 const float* __restrict__ W,
                                 const float* __restrict__ bias,
                                 float* __restrict__ out,
                                 int rowStart, int nRows, int oDim) {
    int idx = blockIdx.x * blockDim.x + threadIdx.x;
    int r = rowStart + idx / oDim;
    int c = idx % oDim;
    if (r >= nRows) return;
    float acc = bias[c];
    for (int k = 0; k < K; ++k) {
        float av = A[(size_t)r * K + k];
        if (RELU) av = fmaxf(av, 0.0f);
        acc += av * W[(size_t)c * K + k];
    }
    out[(size_t)r * oDim + c] = acc;
}

// ===========================================================================
// Gather aggregation (CSR path, atomic-free): one wave per destination node.
//   out[n,:] = dinv[n]^2*(xt[n,:]+be) + sum_{i in seg(n)} dinv[row_i]*dinv[n]
//                                        * (xt[row_i,:] + ea_i*We + be)
// Segment bounds readfirstlane'd so edge metadata becomes SGPR/s_load traffic
// (constant cache), keeping the vector pipe for the 512B feature gathers.
// Edge loop unrolled x2 so two independent gathers are in flight.
// ===========================================================================
template<int DDIM>
__global__ void gather_agg_kernel(const int*   __restrict__ offs,
                                  const int*   __restrict__ permRow,
                                  const float* __restrict__ permEa,
                                  const float* __restrict__ dinv,
                                  const float* __restrict__ xt,
                                  const float* __restrict__ We,
                                  const float* __restrict__ be,
                                  float* __restrict__ out, int nNodes) {
    constexpr int V = DDIM / 32;
    const int lane = threadIdx.x & 31;
    const int wave = threadIdx.x >> 5;
    const int wpb  = blockDim.x >> 5;
    const int gw   = blockIdx.x * wpb + wave;
    const int nW   = gridDim.x * wpb;

    float Wr[V], Br[V];
#pragma unroll
    for (int j = 0; j < V; ++j) { Wr[j] = We[lane * V + j]; Br[j] = be[lane * V + j]; }

    for (int n = gw; n < nNodes; n += nW) {
        const float dn = dinv[n];
        // wave-uniform by construction; make it provable -> scalar loads below
        const int s = __builtin_amdgcn_readfirstlane(offs[n]);
        const int t = __builtin_amdgcn_readfirstlane(offs[n + 1]);
        float acc[V];
        {   // self loop (ea == 0)
            const float* xs = xt + (size_t)n * DDIM + lane * V;
            if constexpr (V == 4) {
                float4 xv = *(const float4*)xs;
                acc[0] = (dn * dn) * (xv.x + Br[0]);
                acc[1] = (dn * dn) * (xv.y + Br[1]);
                acc[2] = (dn * dn) * (xv.z + Br[2]);
                acc[3] = (dn * dn) * (xv.w + Br[3]);
            } else {
                float2 xv = *(const float2*)xs;
                acc[0] = (dn * dn) * (xv.x + Br[0]);
                acc[1] = (dn * dn) * (xv.y + Br[1]);
            }
        }
        int i = s;
        for (; i + 1 < t; i += 2) {                    // 2 gathers in flight
            const int   r0 = permRow[i];
            const int   r1 = permRow[i + 1];
            const float a0 = permEa[i];
            const float a1 = permEa[i + 1];
            const float w0 = dinv[r0] * dn;
            const float w1 = dinv[r1] * dn;
            const float* xp0 = xt + (size_t)r0 * DDIM + lane * V;
            const float* xp1 = xt + (size_t)r1 * DDIM + lane * V;
            if constexpr (V == 4) {
                float4 x0 = *(const float4*)xp0;
                float4 x1 = *(const float4*)xp1;
                acc[0] += w0 * (x0.x + a0 * Wr[0] + Br[0]);
                acc[1] += w0 * (x0.y + a0 * Wr[1] + Br[1]);
                acc[2] += w0 * (x0.z + a0 * Wr[2] + Br[2]);
                acc[3] += w0 * (x0.w + a0 * Wr[3] + Br[3]);
                acc[0] += w1 * (x1.x + a1 * Wr[0] + Br[0]);
                acc[1] += w1 * (x1.y + a1 * Wr[1] + Br[1]);
                acc[2] += w1 * (x1.z + a1 * Wr[2] + Br[2]);
                acc[3] += w1 * (x1.w + a1 * Wr[3] + Br[3]);
            } else {
                float2 x0 = *(const float2*)xp0;
                float2 x1 = *(const float2*)xp1;
                acc[0] += w0 * (x0.x + a0 * Wr[0] + Br[0]);
                acc[1] += w0 * (x0.y + a0 * Wr[1] + Br[1]);
                acc[0] += w1 * (x1.x + a1 * Wr[0] + Br[0]);
                acc[1] += w1 * (x1.y + a1 * Wr[1] + Br[1]);
            }
        }
        if (i < t) {                                   // odd-degree tail
            const int   r  = permRow[i];
            const float a  = permEa[i];
            const float wn = dinv[r] * dn;
            const float* xp = xt + (size_t)r * DDIM + lane * V;
            if constexpr (V == 4) {
                float4 xv = *(const float4*)xp;
                acc[0] += wn * (xv.x + a * Wr[0] + Br[0]);
                acc[1] += wn * (xv.y + a * Wr[1] + Br[1]);
                acc[2] += wn * (xv.z + a * Wr[2] + Br[2]);
                acc[3] += wn * (xv.w + a * Wr[3] + Br[3]);
            } else {
                float2 xv = *(const float2*)xp;
                acc[0] += wn * (xv.x + a * Wr[0] + Br[0]);
                acc[1] += wn * (xv.y + a * Wr[1] + Br[1]);
            }
        }
        float* op = out + (size_t)n * DDIM + lane * V;
        if constexpr (V == 4) *(float4*)op = make_float4(acc[0], acc[1], acc[2], acc[3]);
        else                  *(float2*)op = make_float2(acc[0], acc[1]);
    }
}

// ===========================================================================
// Fallback path (smaller workspace): scatter with f32 hardware atomics.
// ===========================================================================
template<int DDIM>
__global__ void self_init_kernel(const float* __restrict__ xt,
                                 const float* __restrict__ be,
                                 const float* __restrict__ dinv,
                                 float* __restrict__ out, int nNodes) {
    int i = blockIdx.x * blockDim.x + threadIdx.x;
    if (i >= nNodes * DDIM) return;
    int n = i / DDIM;
    int d = i & (DDIM - 1);
    float w = dinv[n];
    out[i] = (w * w) * (xt[i] + be[d]);
}

template<int DDIM>
__global__ void edge_scatter_kernel(const int*   __restrict__ rows,
                                    const int*   __restrict__ cols,
                                    const float* __restrict__ ea,
                                    const float* __restrict__ dinv,
                                    const float* __restrict__ xt,
                                    const float* __restrict__ We,
                                    const float* __restrict__ be,
                                    float* __restrict__ out, int nEdges) {
    constexpr int V = DDIM / 32;
    const int lane = threadIdx.x & 31;
    const int wave = threadIdx.x >> 5;
    const int wpb  = blockDim.x >> 5;
    const int gw   = blockIdx.x * wpb + wave;
    const int nW   = gridDim.x * wpb;

    float Wr[V], Br[V];
#pragma unroll
    for (int j = 0; j < V; ++j) { Wr[j] = We[lane * V + j]; Br[j] = be[lane * V + j]; }

    for (int e = gw; e < nEdges; e += nW) {
        const int r = rows[e];
        const int c = cols[e];
        const float wn = dinv[r] * dinv[c];
        const float a  = ea[e];
        const float* xp = xt  + (size_t)r * DDIM + lane * V;
        float*       op = out + (size_t)c * DDIM + lane * V;
        if constexpr (V == 4) {
            float4 xv = *(const float4*)xp;
            unsafeAtomicAdd(op + 0, wn * (xv.x + a * Wr[0] + Br[0]));
            unsafeAtomicAdd(op + 1, wn * (xv.y + a * Wr[1] + Br[1]));
            unsafeAtomicAdd(op + 2, wn * (xv.z + a * Wr[2] + Br[2]));
            unsafeAtomicAdd(op + 3, wn * (xv.w + a * Wr[3] + Br[3]));
        } else {
            float2 xv = *(const float2*)xp;
            unsafeAtomicAdd(op + 0, wn * (xv.x + a * Wr[0] + Br[0]));
            unsafeAtomicAdd(op + 1, wn * (xv.y + a * Wr[1] + Br[1]));
        }
    }
}

// ===========================================================================
extern "C" void kernel_launch(void* const* d_in, const int* in_sizes, int n_in,
                              void* d_out, int out_size, void* d_ws, size_t ws_size,
                              hipStream_t stream) {
    (void)n_in; (void)out_size;
    const float* x   = (const float*)d_in[0];
    const int*   ei  = (const int*)  d_in[1];
    const float* ea  = (const float*)d_in[2];
    const float* W1  = (const float*)d_in[3];
    const float* b1  = (const float*)d_in[4];
    const float* We1 = (const float*)d_in[5];
    const float* be1 = (const float*)d_in[6];
    const float* W2  = (const float*)d_in[7];
    const float* b2  = (const float*)d_in[8];
    const float* We2 = (const float*)d_in[9];
    const float* be2 = (const float*)d_in[10];
    float* out = (float*)d_out;

    const int N = in_sizes[0] / 128;
    const int E = in_sizes[1] / 2;
    const int* rows = ei;
    const int* cols = ei + E;

    float* w = (float*)d_ws;
    const size_t NP  = (size_t)((N + 3) & ~3);
    const size_t OFP = (size_t)((N + 4) & ~3);     // N+1 entries, padded
    const size_t EP  = (size_t)((E + 3) & ~3);
    const size_t FEAT = (size_t)N * (128 + 128 + 64);
    const size_t need_csr = sizeof(float) * (NP + OFP + NP + 2 * EP + FEAT);

    const int T = 256;
    const int mTiles = N / 16;
    const int rem = N - mTiles * 16;

    if (ws_size >= need_csr) {
        // ------------------ CSR (gather, atomic-free) path ------------------
        float* dinv    = w;
        int*   offs    = (int*)(w + NP);
        int*   cursor  = (int*)(w + NP + OFP);
        int*   permRow = (int*)(w + NP + OFP + NP);
        float* permEa  = (float*)(w + NP + OFP + NP + EP);
        float* xt      = w + NP + OFP + NP + 2 * EP;
        float* h       = xt + (size_t)N * 128;
        float* xt2     = h  + (size_t)N * 128;

        zero_int_kernel<<<(N + T - 1) / T, T, 0, stream>>>(cursor, N);
        count_kernel   <<<(E + T - 1) / T, T, 0, stream>>>(cols, cursor, E);
        scan_kernel    <<<1, 1024, 0, stream>>>(cursor, offs, N);
        zero_int_kernel<<<(N + T - 1) / T, T, 0, stream>>>(cursor, N);
        permute_kernel <<<(E + T - 1) / T, T, 0, stream>>>(rows, cols, ea, offs, cursor,
                                                           permRow, permEa, E);
        dinv_from_offs_kernel<<<(N + T - 1) / T, T, 0, stream>>>(offs, dinv, N);

        gemm_bias_wmma<128, 128, false><<<(mTiles * 8 + 7) / 8, T, 0, stream>>>(x, W1, b1, xt, mTiles);
        if (rem) gemm_tail_kernel<128, false><<<(rem * 128 + T - 1) / T, T, 0, stream>>>(
                     x, W1, b1, xt, mTiles * 16, N, 128);
        gather_agg_kernel<128><<<(N + 7) / 8, T, 0, stream>>>(offs, permRow, permEa, dinv,
                                                              xt, We1, be1, h, N);

        gemm_bias_wmma<128, 64, true><<<(mTiles * 4 + 7) / 8, T, 0, stream>>>(h, W2, b2, xt2, mTiles);
        if (rem) gemm_tail_kernel<128, true><<<(rem * 64 + T - 1) / T, T, 0, stream>>>(
                     h, W2, b2, xt2, mTiles * 16, N, 64);
        gather_agg_kernel<64><<<(N + 7) / 8, T, 0, stream>>>(offs, permRow, permEa, dinv,
                                                             xt2, We2, be2, out, N);
    } else {
        // ------------------ fallback scatter (atomic) path ------------------
        float* dinv = w;
        float* xt   = w + NP;
        float* h    = xt + (size_t)N * 128;
        float* xt2  = h  + (size_t)N * 128;

        deg_init_kernel  <<<(N + T - 1) / T, T, 0, stream>>>(dinv, N);
        deg_accum_kernel <<<(E + T - 1) / T, T, 0, stream>>>(cols, dinv, E);
        deg_finish_kernel<<<(N + T - 1) / T, T, 0, stream>>>(dinv, N);

        gemm_bias_wmma<128, 128, false><<<(mTiles * 8 + 7) / 8, T, 0, stream>>>(x, W1, b1, xt, mTiles);
        if (rem) gemm_tail_kernel<128, false><<<(rem * 128 + T - 1) / T, T, 0, stream>>>(
                     x, W1, b1, xt, mTiles * 16, N, 128);
        self_init_kernel<128><<<((long long)N * 128 + T - 1) / T, T, 0, stream>>>(xt, be1, dinv, h, N);
        edge_scatter_kernel<128><<<4096, T, 0, stream>>>(rows, cols, ea, dinv, xt, We1, be1, h, E);

        gemm_bias_wmma<128, 64, true><<<(mTiles * 4 + 7) / 8, T, 0, stream>>>(h, W2, b2, xt2, mTiles);
        if (rem) gemm_tail_kernel<128, true><<<(rem * 64 + T - 1) / T, T, 0, stream>>>(
                     h, W2, b2, xt2, mTiles * 16, N, 64);
        self_init_kernel<64><<<((long long)N * 64 + T - 1) / T, T, 0, stream>>>(xt2, be2, dinv, out, N);
        edge_scatter_kernel<64><<<4096, T, 0, stream>>>(rows, cols, ea, dinv, xt2, We2, be2, out, E);
    }
}